// RBF_PINNs_56057913147760
// MI455X (gfx1250) — compile-verified
//
#include <hip/hip_runtime.h>
#include <math.h>

// D = A(16x4 f32) x B(4x16 f32) + C(16x16 f32), wave32.
typedef __attribute__((ext_vector_type(2))) float v2f;
typedef __attribute__((ext_vector_type(8))) float v8f;

#define UNITS 256
#define TPB 256
#define WPB (TPB / 32)   // waves per block
#define NBLOCKS 256
#define LOG2E 1.4426950408889634f

// ---------------------------------------------------------------------------
// Kernel 0: zero the workspace accumulators (d_ws is poisoned, not zeroed).
__global__ void k_init_ws(float* __restrict__ ws) {
    if (threadIdx.x < 4) ws[threadIdx.x] = 0.0f;
}

// ---------------------------------------------------------------------------
// Kernel 1: fused RBF-PINN evaluation.
//   phi_ij = exp(-0.5*r2) = 2^(-0.5*rho*L) * 2^(x*cx' + y*cy' + h')   (L=log2 e)
//   Moments per point i:  M_k(i) = sum_j phi_ij * (w_j * f_k(center_j))
//   computed as GEMM Psi(16x256) x V(256x16) via V_WMMA_F32_16X16X4_F32,
//   with the point-only factor 2^(-0.5*rho*L) applied at finalize.
//   f = {1, cx, cy, c2, cx^2, cy^2, cx*cy, cx*c2, cy*c2, c2^2, 0...}
__global__ __launch_bounds__(TPB) void k_pinn_main(
    const float* __restrict__ x, const float* __restrict__ y,
    const float* __restrict__ utgt, const float* __restrict__ centers,
    const float* __restrict__ W, const float* __restrict__ eps_p,
    const float* __restrict__ delta_p, const float* __restrict__ gamma_p,
    const int* __restrict__ iter_p, float* __restrict__ ws, int N)
{
    __shared__ float Vl[UNITS * 16];      // 16 KB: V matrix (row-major, 16 cols)
    __shared__ float cxl[UNITS];          // cx * log2(e)
    __shared__ float cyl[UNITS];          // cy * log2(e)
    __shared__ float hcl[UNITS];          // -0.5 * (cx^2+cy^2) * log2(e)
    __shared__ float tiles[WPB * 256];    // per-wave 16x16 C-tile spill

    const int tid = threadIdx.x;

    // Build V and exponent tables (TPB == UNITS: one thread per center).
    {
        const int j = tid;
        const float w  = W[j];
        const float cx = centers[2 * j + 0];
        const float cy = centers[2 * j + 1];
        const float c2 = cx * cx + cy * cy;
        cxl[j] = cx * LOG2E;
        cyl[j] = cy * LOG2E;
        hcl[j] = -0.5f * c2 * LOG2E;
        float* row = &Vl[j * 16];
        row[0] = w;            row[1] = w * cx;       row[2] = w * cy;
        row[3] = w * c2;       row[4] = w * cx * cx;  row[5] = w * cy * cy;
        row[6] = w * cx * cy;  row[7] = w * cx * c2;  row[8] = w * cy * c2;
        row[9] = w * c2 * c2;
        row[10] = 0.f; row[11] = 0.f; row[12] = 0.f;
        row[13] = 0.f; row[14] = 0.f; row[15] = 0.f;
    }
    __syncthreads();

    const float eps   = eps_p[0];
    const float delta = delta_p[0];
    const float gamma = gamma_p[0];
    const int   it    = iter_p[0];
    float ca, cb;                          // residual nonlinearity coefficients
    if      (it < 10000) { ca = 0.0f; cb = 0.0f; }
    else if (it < 20000) { ca = 0.3f; cb = 0.0f; }
    else if (it < 30000) { ca = 0.7f; cb = 0.3f; }
    else                 { ca = 1.0f; cb = 1.0f; }
    const float ad = ca * delta;
    const float bg = cb * gamma;

    const int lane       = tid & 31;
    const int waveInBlk  = tid >> 5;
    const int wave       = blockIdx.x * WPB + waveInBlk;
    const int totalWaves = gridDim.x * WPB;
    const int nTiles     = N >> 4;         // 16 points per tile
    const int m          = lane & 15;      // A-matrix row (point within tile)
    const int khalf      = (lane < 16) ? 0 : 2;   // K sub-pair per ISA layout
    float* tileL = &tiles[waveInBlk * 256];

    float acc1 = 0.f, acc2 = 0.f, acc3 = 0.f;

    for (int t = wave; t < nTiles; t += totalWaves) {
        const int p  = t * 16 + m;
        const float xp = x[p];
        const float yp = y[p];

        v8f c = {0.f, 0.f, 0.f, 0.f, 0.f, 0.f, 0.f, 0.f};
        #pragma unroll 4
        for (int kk = 0; kk < 64; ++kk) {
            const int k0 = kk * 4 + khalf;
            // psi = 2^(x*cx' + y*cy' + h')  : 2 fma + 1 v_exp per element
            const float e0 = __builtin_fmaf(xp, cxl[k0],
                             __builtin_fmaf(yp, cyl[k0],     hcl[k0]));
            const float e1 = __builtin_fmaf(xp, cxl[k0 + 1],
                             __builtin_fmaf(yp, cyl[k0 + 1], hcl[k0 + 1]));
            const float p0 = __builtin_amdgcn_exp2f(e0);
            const float p1 = __builtin_amdgcn_exp2f(e1);
            v2f a = {p0, p1};                                   // A frag (2 VGPRs)
            v2f b = {Vl[k0 * 16 + m], Vl[(k0 + 1) * 16 + m]};   // B frag (2 VGPRs)
            // (neg_a, A, neg_b, B, c_mod, C, reuse_a, reuse_b)
            c = __builtin_amdgcn_wmma_f32_16x16x4_f32(
                    false, a, false, b, (short)0, c, false, false);
        }

        // Spill 16x16 moment tile to LDS per ISA C-layout:
        // VGPR v, lanes 0-15 -> row v; lanes 16-31 -> row v+8; col = lane%16.
        const int hi = (lane < 16) ? 0 : 8;
        #pragma unroll
        for (int v = 0; v < 8; ++v)
            tileL[(v + hi) * 16 + m] = c[v];

        if (lane < 16) {   // re-converges before next WMMA (EXEC full there)
            float M[10];
            #pragma unroll
            for (int n = 0; n < 10; ++n) M[n] = tileL[lane * 16 + n];

            const float uq  = utgt[p];
            const float rho = xp * xp + yp * yp;
            // Point-only Gaussian factor pulled out of the K-loop:
            const float ep  = __builtin_amdgcn_exp2f(-0.5f * LOG2E * rho);

            const float u0 = ep * M[0];
            // S1 = sum w*phi*r^2 ; S2 = sum w*phi*r^4
            const float S1 = ep * (rho * M[0] - 2.f * xp * M[1]
                                   - 2.f * yp * M[2] + M[3]);
            const float S2 = ep * (rho * rho * M[0] + M[9]
                                   + 4.f * xp * xp * M[4] + 4.f * yp * yp * M[5]
                                   + 2.f * rho * M[3]
                                   - 4.f * rho * xp * M[1] - 4.f * rho * yp * M[2]
                                   - 4.f * xp * M[7] - 4.f * yp * M[8]
                                   + 8.f * xp * yp * M[6]);
            // u + 2*lap + bih  with lap = S1-2S0, bih = S2-8S1+8S0
            const float sh  = 5.f * u0 - 6.f * S1 + S2;
            const float u2  = u0 * u0;
            const float res = eps * u0 - ad * u2 - bg * u2 * u0 - sh;
            const float du  = u0 - uq;
            acc1 += du * du;
            acc2 += uq * uq;
            acc3 += res * res;
        }
    }

    // Full-wave reduction (lanes 16..31 carry zeros).
    for (int off = 16; off > 0; off >>= 1) {
        acc1 += __shfl_down(acc1, off, 32);
        acc2 += __shfl_down(acc2, off, 32);
        acc3 += __shfl_down(acc3, off, 32);
    }
    if (lane == 0) {
        atomicAdd(&ws[0], acc1);
        atomicAdd(&ws[1], acc2);
        atomicAdd(&ws[2], acc3);
    }
}

// ---------------------------------------------------------------------------
// Kernel 2: finalize the three scalar losses.
__global__ void k_finalize(const float* __restrict__ ws,
                           const float* __restrict__ eps_p,
                           const int* __restrict__ iter_p,
                           float* __restrict__ out, int N)
{
    if (threadIdx.x == 0) {
        const float loss_u   = ws[0] / ws[1];
        const float loss_pde = ws[2] / (float)N;
        const float eps = eps_p[0];
        const int   it  = iter_p[0];
        const float cw  = (it < 10000) ? 0.1f
                        : (it < 20000) ? 0.3f
                        : (it < 30000) ? 0.6f : 1.0f;
        const float a = fmaxf(0.f, eps - 0.8f);
        const float b = fmaxf(0.f, 0.2f - eps);
        const float eps_bp = a * a + b * b;
        out[0] = 0.1f * loss_u + cw * loss_pde + 0.01f * eps_bp;
        out[1] = loss_u;
        out[2] = loss_pde;
    }
}

// ---------------------------------------------------------------------------
extern "C" void kernel_launch(void* const* d_in, const int* in_sizes, int n_in,
                              void* d_out, int out_size, void* d_ws, size_t ws_size,
                              hipStream_t stream)
{
    const float* x       = (const float*)d_in[0];
    const float* y       = (const float*)d_in[1];
    const float* u       = (const float*)d_in[2];
    const float* centers = (const float*)d_in[3];
    const float* W       = (const float*)d_in[4];
    const float* eps_p   = (const float*)d_in[5];
    const float* delta_p = (const float*)d_in[6];
    const float* gamma_p = (const float*)d_in[7];
    const int*   iter_p  = (const int*)d_in[8];
    float* out = (float*)d_out;
    float* ws  = (float*)d_ws;
    const int N = in_sizes[0];

    k_init_ws<<<1, 32, 0, stream>>>(ws);
    k_pinn_main<<<NBLOCKS, TPB, 0, stream>>>(x, y, u, centers, W, eps_p,
                                             delta_p, gamma_p, iter_p, ws, N);
    k_finalize<<<1, 1, 0, stream>>>(ws, eps_p, iter_p, out, N);
}